// HeteroGNN_25537875542278
// MI455X (gfx1250) — compile-verified
//
#include <hip/hip_runtime.h>

// Heterogeneous 2-layer GraphSAGE (D=32, f32) for MI455X / gfx1250.
// Sparse phase: wave-per-edge gather + GLOBAL_ATOMIC_ADD_F32 segment sums.
// Dense phase: fused [x_dst | mean] @ [Ws^T ; Wn^T] via V_WMMA_F32_16X16X4_F32.

#define NF 32           // feature dim D
#define N_USER 200000
#define N_ITEM 200000

typedef __attribute__((ext_vector_type(2))) float v2f;
typedef __attribute__((ext_vector_type(8))) float v8f;

// ---------------------------------------------------------------------------
// Sparse scatter: one wave32 per edge. Coalesced 128B row gather, 32 float
// atomics to the destination row, lane 0 bumps the degree counter.
// ---------------------------------------------------------------------------
__global__ void __launch_bounds__(256)
hgnn_scatter_kernel(const float* __restrict__ xsrc,
                    const int* __restrict__ src,
                    const int* __restrict__ dst,
                    float* __restrict__ sum,
                    float* __restrict__ deg,
                    int E) {
  const int lane = threadIdx.x & 31;
  int wid = (int)((blockIdx.x * blockDim.x + threadIdx.x) >> 5);
  const int nw = (int)((gridDim.x * blockDim.x) >> 5);
  for (int e = wid; e < E; e += nw) {
    const int s = src[e];
    const int d = dst[e];
    const float v = xsrc[(size_t)s * NF + lane];
    atomicAdd(&sum[(size_t)d * NF + lane], v);
    if (lane == 0) atomicAdd(&deg[d], 1.0f);
  }
}

// ---------------------------------------------------------------------------
// Dense SAGE transform, one wave32 per 16-node tile:
//   out[m][n] = bias[n] + sum_{k<32} x[m][k]*Ws[n][k]
//                       + sum_{k<32} mean[m][k]*Wn[n][k]
// Expressed as C(16x32) = A(16x64) x B(64x32) + bias with
// A = [x | sum*inv_deg], B = [Ws^T ; Wn^T], via 16x16x4 f32 WMMA.
//
// f32 WMMA VGPR layouts (ISA 7.12.2), lane l, reg r:
//   A 16x4 : M = l&15,           K = (l>>4)*2 + r
//   B 4x16 : N = l&15,           K = (l>>4)*2 + r
//   C 16x16: N = l&15,           M = (l>>4)*8 + r   (r = 0..7)
// ---------------------------------------------------------------------------
__global__ void __launch_bounds__(256)
hgnn_sage_wmma_kernel(const float* __restrict__ xdst,
                      const float* __restrict__ ssum,
                      const float* __restrict__ deg,
                      const float* __restrict__ Ws,
                      const float* __restrict__ Wn,
                      const float* __restrict__ bias,
                      float* __restrict__ out,
                      int n, int do_relu) {
  const int lane = threadIdx.x & 31;
  const int wave = threadIdx.x >> 5;
  const int tile = blockIdx.x * (blockDim.x >> 5) + wave;
  const int ntiles = (n + 15) >> 4;
  if (tile >= ntiles) return;           // whole-wave exit: EXEC all-1s inside

  const int lg = lane >> 4;             // lane group 0/1
  const int lm = lane & 15;             // lane-minor

  // ---- B fragments: stacked weights, K = 0..63, N = 0..31 ----------------
  // bfrag[s][h][r] = B[K = 4s + 2*lg + r][N = 16h + lm]
  v2f bfrag[16][2];
#pragma unroll
  for (int s = 0; s < 16; ++s) {
#pragma unroll
    for (int h = 0; h < 2; ++h) {
#pragma unroll
      for (int r = 0; r < 2; ++r) {
        const int k = 4 * s + 2 * lg + r;
        const int nn = 16 * h + lm;
        const float w = (k < NF) ? Ws[nn * NF + k] : Wn[nn * NF + (k - NF)];
        bfrag[s][h][r] = w;
      }
    }
  }

  // ---- A fragments: [x_dst | sum/max(deg,1)] ------------------------------
  const int row0 = tile << 4;
  int arow = row0 + lm;
  if (arow > n - 1) arow = n - 1;       // clamp (tail-safe; 200000 % 16 == 0)
  const float invd = 1.0f / fmaxf(deg[arow], 1.0f);
  const float* __restrict__ xr = xdst + (size_t)arow * NF;
  const float* __restrict__ sr = ssum + (size_t)arow * NF;

  v2f afrag[16];
#pragma unroll
  for (int s = 0; s < 16; ++s) {
#pragma unroll
    for (int r = 0; r < 2; ++r) {
      const int k = 4 * s + 2 * lg + r;
      afrag[s][r] = (k < NF) ? xr[k] : sr[k - NF] * invd;
    }
  }

  // ---- Accumulators seeded with bias (bias depends only on N) -------------
  const float b0 = bias[lm];
  const float b1 = bias[16 + lm];
  v8f acc0, acc1;
#pragma unroll
  for (int r = 0; r < 8; ++r) { acc0[r] = b0; acc1[r] = b1; }

  // ---- 16 K-steps x 2 N-halves = 32 WMMAs ---------------------------------
#pragma unroll
  for (int s = 0; s < 16; ++s) {
    acc0 = __builtin_amdgcn_wmma_f32_16x16x4_f32(
        false, afrag[s], false, bfrag[s][0], (short)0, acc0, false, false);
    acc1 = __builtin_amdgcn_wmma_f32_16x16x4_f32(
        false, afrag[s], false, bfrag[s][1], (short)0, acc1, false, false);
  }

  // ---- Store D: lane l, reg r -> row row0 + 8*lg + r, col lm / lm+16 ------
#pragma unroll
  for (int r = 0; r < 8; ++r) {
    const int m = row0 + 8 * lg + r;
    if (m < n) {
      float v0 = acc0[r];
      float v1 = acc1[r];
      if (do_relu) { v0 = fmaxf(v0, 0.0f); v1 = fmaxf(v1, 0.0f); }
      out[(size_t)m * NF + lm]      = v0;
      out[(size_t)m * NF + 16 + lm] = v1;
    }
  }
}

// ---------------------------------------------------------------------------
extern "C" void kernel_launch(void* const* d_in, const int* in_sizes, int n_in,
                              void* d_out, int out_size, void* d_ws, size_t ws_size,
                              hipStream_t stream) {
  const float* x_user = (const float*)d_in[0];
  const float* x_item = (const float*)d_in[1];
  const int* e_ui_src = (const int*)d_in[2];
  const int* e_ui_dst = (const int*)d_in[3];
  const int* e_iu_src = (const int*)d_in[4];
  const int* e_iu_dst = (const int*)d_in[5];
  const float* Ws_ui1 = (const float*)d_in[6];
  const float* Wn_ui1 = (const float*)d_in[7];
  const float* Ws_iu1 = (const float*)d_in[8];
  const float* Wn_iu1 = (const float*)d_in[9];
  const float* Ws_ui2 = (const float*)d_in[10];
  const float* Wn_ui2 = (const float*)d_in[11];
  const float* Ws_iu2 = (const float*)d_in[12];
  const float* Wn_iu2 = (const float*)d_in[13];
  const float* b_ui1  = (const float*)d_in[14];
  const float* b_iu1  = (const float*)d_in[15];
  const float* b_ui2  = (const float*)d_in[16];
  const float* b_iu2  = (const float*)d_in[17];

  const int E = in_sizes[2];

  // Workspace layout (floats), accumulators first so one memset clears them.
  float* ws = (float*)d_ws;
  float* sum_item = ws;                               // N_ITEM*32
  float* sum_user = sum_item + (size_t)N_ITEM * NF;   // N_USER*32
  float* deg_item = sum_user + (size_t)N_USER * NF;   // N_ITEM
  float* deg_user = deg_item + N_ITEM;                // N_USER
  float* h_item   = deg_user + N_USER;                // N_ITEM*32
  float* h_user   = h_item + (size_t)N_ITEM * NF;     // N_USER*32
  const size_t accum_bytes =
      ((size_t)N_ITEM * NF + (size_t)N_USER * NF + N_ITEM + N_USER) * sizeof(float);

  float* o_user = (float*)d_out;                      // tuple order: (o_user, o_item)
  float* o_item = o_user + (size_t)N_USER * NF;

  const dim3 blk(256);
  const dim3 scat_grid(4096);                          // grid-stride, 32768 waves
  const int tiles_u = (N_USER + 15) / 16;
  const int tiles_i = (N_ITEM + 15) / 16;
  const dim3 sage_grid_u((tiles_u + 7) / 8);           // 8 waves per block
  const dim3 sage_grid_i((tiles_i + 7) / 8);

  // ---------------- Layer 1 ----------------
  hipMemsetAsync(ws, 0, accum_bytes, stream);
  hgnn_scatter_kernel<<<scat_grid, blk, 0, stream>>>(x_user, e_ui_src, e_ui_dst,
                                                     sum_item, deg_item, E);
  hgnn_scatter_kernel<<<scat_grid, blk, 0, stream>>>(x_item, e_iu_src, e_iu_dst,
                                                     sum_user, deg_user, E);
  hgnn_sage_wmma_kernel<<<sage_grid_i, blk, 0, stream>>>(
      x_item, sum_item, deg_item, Ws_ui1, Wn_ui1, b_ui1, h_item, N_ITEM, 1);
  hgnn_sage_wmma_kernel<<<sage_grid_u, blk, 0, stream>>>(
      x_user, sum_user, deg_user, Ws_iu1, Wn_iu1, b_iu1, h_user, N_USER, 1);

  // ---------------- Layer 2 ----------------
  hipMemsetAsync(ws, 0, accum_bytes, stream);
  hgnn_scatter_kernel<<<scat_grid, blk, 0, stream>>>(h_user, e_ui_src, e_ui_dst,
                                                     sum_item, deg_item, E);
  hgnn_scatter_kernel<<<scat_grid, blk, 0, stream>>>(h_item, e_iu_src, e_iu_dst,
                                                     sum_user, deg_user, E);
  hgnn_sage_wmma_kernel<<<sage_grid_i, blk, 0, stream>>>(
      h_item, sum_item, deg_item, Ws_ui2, Wn_ui2, b_ui2, o_item, N_ITEM, 0);
  hgnn_sage_wmma_kernel<<<sage_grid_u, blk, 0, stream>>>(
      h_user, sum_user, deg_user, Ws_iu2, Wn_iu2, b_iu2, o_user, N_USER, 0);
}